// GatingFunction_68650757260117
// MI455X (gfx1250) — compile-verified
//
#include <hip/hip_runtime.h>

typedef __attribute__((ext_vector_type(2))) float v2f;
typedef __attribute__((ext_vector_type(8))) float v8f;
typedef __attribute__((ext_vector_type(4))) int   v4i;

#define D_MODEL        1024
#define NEXPERT        8
#define TPB            256
#define TOK_PER_BLOCK  128
#define KTILE          16
#define NTILE          (D_MODEL / KTILE)     // 64
#define XSTRIDE        (KTILE + 4)           // 20 floats -> conflict-free A reads
#define XBUF           (TOK_PER_BLOCK * XSTRIDE)
#define WSTRIDE        (D_MODEL + 4)         // 1028 floats -> conflict-free B reads
#define LSTRIDE        (NEXPERT + 1)         // 9

#if defined(__has_builtin)
#if __has_builtin(__builtin_amdgcn_global_load_async_to_lds_b128)
#define USE_ASYNC_LDS 1
#endif
#if __has_builtin(__builtin_amdgcn_s_wait_asynccnt)
#define HAVE_WAIT_ASYNC_BUILTIN 1
#endif
#endif

__device__ __forceinline__ void async_copy_b128(const float* g, float* l) {
#ifdef USE_ASYNC_LDS
    // builtin signature (from hipcc diagnostic): arg0 = v4i AS(1)* (global src),
    // arg1 = lds dst, then imm offset + imm cpol
    __builtin_amdgcn_global_load_async_to_lds_b128(
        (__attribute__((address_space(1))) v4i*)g,
        (__attribute__((address_space(3))) v4i*)l, 0, 0);
#else
    *(float4*)l = *(const float4*)g;   // fallback: sync load + ds_store
#endif
}

__device__ __forceinline__ void wait_async0() {
#ifdef USE_ASYNC_LDS
#ifdef HAVE_WAIT_ASYNC_BUILTIN
    __builtin_amdgcn_s_wait_asynccnt(0);
#else
    asm volatile("s_wait_asynccnt 0x0" ::: "memory");
#endif
#endif
}

__global__ __launch_bounds__(TPB) void moe_gate_wmma_kernel(
    const float* __restrict__ x, const float* __restrict__ W,
    const float* __restrict__ bias, float* __restrict__ w_out,
    int* __restrict__ idx_out)
{
    __shared__ float sW[NEXPERT * WSTRIDE];
    __shared__ float sX[2 * XBUF];
    __shared__ float sB[NEXPERT];
    __shared__ float sL[TOK_PER_BLOCK * LSTRIDE];

    const int tid    = threadIdx.x;
    const int wave   = tid >> 5;
    const int lane   = tid & 31;
    const int n      = lane & 15;            // expert column (8..15 = harmless dups)
    const int kbase  = (lane < 16) ? 0 : 2;  // K split per ISA A/B fragment layout
    const int nclamp = n & 7;

    const long tok_base = (long)blockIdx.x * TOK_PER_BLOCK;

    // ---- stage W and bias into LDS once (coalesced) ----
    for (int e = 0; e < NEXPERT; ++e) {
#pragma unroll
        for (int j = 0; j < D_MODEL / TPB; ++j) {
            int c = tid + j * TPB;
            sW[e * WSTRIDE + c] = W[e * D_MODEL + c];
        }
    }
    if (tid < NEXPERT) sB[tid] = bias[tid];

    // ---- async x-tile staging, double buffered ----
    const int arow = tid >> 2;          // 0..63 (4 threads per token row)
    const int acol = (tid & 3) * 4;     // 0,4,8,12

    auto issue_tile = [&](int kt, int buf) {
#pragma unroll
        for (int r = 0; r < TOK_PER_BLOCK / 64; ++r) {
            const int row = arow + r * 64;
            async_copy_b128(&x[(tok_base + row) * D_MODEL + kt * KTILE + acol],
                            &sX[buf * XBUF + row * XSTRIDE + acol]);
        }
    };

    issue_tile(0, 0);                   // prologue: tile 0 -> buffer 0

    v8f acc = {};
    const int xoff = (wave * 16 + (lane & 15)) * XSTRIDE;

    for (int kt = 0; kt < NTILE; ++kt) {
        wait_async0();                  // my tile-kt loads have landed in LDS
        __syncthreads();                // everyone's have; old buffer fully consumed
        if (kt + 1 < NTILE) issue_tile(kt + 1, (kt + 1) & 1);

        const float* xrow = &sX[(kt & 1) * XBUF + xoff];
        const float* wrow = &sW[nclamp * WSTRIDE + kt * KTILE];
#pragma unroll
        for (int kk = 0; kk < KTILE; kk += 4) {
            const v2f a  = *(const v2f*)&xrow[kk + kbase];
            const v2f bf = *(const v2f*)&wrow[kk + kbase];
            acc = __builtin_amdgcn_wmma_f32_16x16x4_f32(
                false, a, false, bf, (short)0, acc, false, false);
        }
    }

    // ---- scatter logits (+bias) through LDS ----
#pragma unroll
    for (int r = 0; r < 8; ++r) {
        int m = r + ((lane < 16) ? 0 : 8);
        if (n < NEXPERT)
            sL[(wave * 16 + m) * LSTRIDE + n] = acc[r] + sB[n];
    }
    __syncthreads();

    // ---- per-token top-2 + softmax + scatter ----
    if (tid < TOK_PER_BLOCK) {
        float lg[NEXPERT];
#pragma unroll
        for (int e = 0; e < NEXPERT; ++e) lg[e] = sL[tid * LSTRIDE + e];

        int bi = 0; float bv = lg[0];
#pragma unroll
        for (int e = 1; e < NEXPERT; ++e)
            if (lg[e] > bv) { bv = lg[e]; bi = e; }   // strict > == top_k tiebreak

        int si = 0; float sv = -3.402823466e38f;
#pragma unroll
        for (int e = 0; e < NEXPERT; ++e)
            if (e != bi && lg[e] > sv) { sv = lg[e]; si = e; }

        float t  = __expf(sv - bv);
        float w0 = 1.0f / (1.0f + t);
        float w1 = t * w0;

        const long gt = tok_base + tid;
        float wv[NEXPERT];
#pragma unroll
        for (int e = 0; e < NEXPERT; ++e)
            wv[e] = (e == bi) ? w0 : ((e == si) ? w1 : 0.0f);
        *(float4*)&w_out[gt * NEXPERT + 0] = make_float4(wv[0], wv[1], wv[2], wv[3]);
        *(float4*)&w_out[gt * NEXPERT + 4] = make_float4(wv[4], wv[5], wv[6], wv[7]);
        idx_out[gt * 2 + 0] = bi;
        idx_out[gt * 2 + 1] = si;
    }
}

extern "C" void kernel_launch(void* const* d_in, const int* in_sizes, int n_in,
                              void* d_out, int out_size, void* d_ws, size_t ws_size,
                              hipStream_t stream) {
    const float* x = (const float*)d_in[0];
    const float* W = (const float*)d_in[1];
    const float* b = (const float*)d_in[2];
    const int n_tokens = in_sizes[0] / D_MODEL;            // 32768
    float* wout = (float*)d_out;                           // [n_tokens, 8] f32
    int*   iout = (int*)(wout + (long)n_tokens * NEXPERT); // [n_tokens, 2] i32
    const int grid = n_tokens / TOK_PER_BLOCK;             // 256
    moe_gate_wmma_kernel<<<grid, TPB, 0, stream>>>(x, W, b, wout, iout);
}